// AttnDecoderRNN_49813030699682
// MI455X (gfx1250) — compile-verified
//
#include <hip/hip_runtime.h>
#include <math.h>

typedef float v2f __attribute__((ext_vector_type(2)));
typedef float v8f __attribute__((ext_vector_type(8)));

#define H 1024
#define ML 4096
#define V 128000

// ---------- build xcat1 = [e0, h0]; xcat2[0:H] = e0 ----------
__global__ void prep_kernel(const int* __restrict__ ids, const float* __restrict__ hidden,
                            const float* __restrict__ emb,
                            float* __restrict__ xcat1, float* __restrict__ xcat2) {
  int i = blockIdx.x * blockDim.x + threadIdx.x;  // 0..2047
  int id = ids[0];
  if (i < H) {
    float e = emb[(size_t)id * H + i];
    xcat1[i] = e;
    xcat2[i] = e;
  } else if (i < 2 * H) {
    xcat1[i] = hidden[i - H];
  }
}

// ---------- generic WMMA GEMV: y[r] = dot(W[r,:], x) + bias[r], optional ReLU ----------
// One wave computes 16 rows via V_WMMA_F32_16X16X4_F32.
// A fragment (16x4 f32): lanes 0-15 hold M=0..15 K0/K1, lanes 16-31 hold K2/K3.
// B fragment (4x16 f32): x chunk broadcast across all 16 N columns -> every column
// of the 16x16 fp32 accumulator equals the desired dot product (column 0 is read out).
template <bool NT>
__global__ void wmma_gemv(const float* __restrict__ W, const float* __restrict__ x,
                          const float* __restrict__ bias, float* __restrict__ y,
                          int K, int act) {
  __shared__ float xs[2048];
  for (int i = threadIdx.x; i < K; i += blockDim.x) xs[i] = x[i];
  __syncthreads();

  const int lane = threadIdx.x & 31;
  const int wave = threadIdx.x >> 5;
  const int row0 = blockIdx.x * 128 + wave * 16;
  const int m    = lane & 15;
  const int hi   = lane >> 4;  // 0: K0/K1 half, 1: K2/K3 half
  const float* wrow = W + (size_t)(row0 + m) * (size_t)K + (size_t)(2 * hi);

  v8f acc0 = {0.f,0.f,0.f,0.f,0.f,0.f,0.f,0.f};
  v8f acc1 = acc0, acc2 = acc0, acc3 = acc0;

  for (int k0 = 0; k0 < K; k0 += 16) {
    v2f a0, a1, a2, a3;
    if (NT) {
      a0 = __builtin_nontemporal_load((const v2f*)(wrow + k0));
      a1 = __builtin_nontemporal_load((const v2f*)(wrow + k0 + 4));
      a2 = __builtin_nontemporal_load((const v2f*)(wrow + k0 + 8));
      a3 = __builtin_nontemporal_load((const v2f*)(wrow + k0 + 12));
    } else {
      a0 = *(const v2f*)(wrow + k0);
      a1 = *(const v2f*)(wrow + k0 + 4);
      a2 = *(const v2f*)(wrow + k0 + 8);
      a3 = *(const v2f*)(wrow + k0 + 12);
    }
    v2f b0, b1, b2, b3;
    b0.x = xs[k0 +      2 * hi]; b0.y = xs[k0 +      2 * hi + 1];
    b1.x = xs[k0 + 4  + 2 * hi]; b1.y = xs[k0 + 4  + 2 * hi + 1];
    b2.x = xs[k0 + 8  + 2 * hi]; b2.y = xs[k0 + 8  + 2 * hi + 1];
    b3.x = xs[k0 + 12 + 2 * hi]; b3.y = xs[k0 + 12 + 2 * hi + 1];

    acc0 = __builtin_amdgcn_wmma_f32_16x16x4_f32(false, a0, false, b0, (short)0, acc0, false, false);
    acc1 = __builtin_amdgcn_wmma_f32_16x16x4_f32(false, a1, false, b1, (short)0, acc1, false, false);
    acc2 = __builtin_amdgcn_wmma_f32_16x16x4_f32(false, a2, false, b2, (short)0, acc2, false, false);
    acc3 = __builtin_amdgcn_wmma_f32_16x16x4_f32(false, a3, false, b3, (short)0, acc3, false, false);
  }

  v8f acc = (acc0 + acc1) + (acc2 + acc3);

  // C/D layout: lane n in 0-15 -> N=n rows M=0..7 in VGPR 0..7; lane 16+n -> rows 8..15.
  // B was broadcast across N, so lanes 0 and 16 (N=0) hold the 16 row results.
  if (m == 0) {
    int base = row0 + hi * 8;
#pragma unroll
    for (int g = 0; g < 8; ++g) {
      float v = acc[g] + bias[base + g];
      if (act) v = fmaxf(v, 0.0f);
      y[base + g] = v;
    }
  }
}

// ---------- softmax over ML logits; also zero the attn accumulator ----------
__global__ void softmax_kernel(const float* __restrict__ logits, float* __restrict__ wout,
                               float* __restrict__ attn_acc) {
  __shared__ float red[1024];
  int tid = threadIdx.x;
  float mx = -1e30f;
  for (int i = tid; i < ML; i += 1024) mx = fmaxf(mx, logits[i]);
  red[tid] = mx; __syncthreads();
  for (int s = 512; s > 0; s >>= 1) {
    if (tid < s) red[tid] = fmaxf(red[tid], red[tid + s]);
    __syncthreads();
  }
  mx = red[0]; __syncthreads();

  float sum = 0.f;
  for (int i = tid; i < ML; i += 1024) sum += expf(logits[i] - mx);
  red[tid] = sum; __syncthreads();
  for (int s = 512; s > 0; s >>= 1) {
    if (tid < s) red[tid] += red[tid + s];
    __syncthreads();
  }
  sum = red[0]; __syncthreads();

  float inv = 1.0f / sum;
  for (int i = tid; i < ML; i += 1024) wout[i] = expf(logits[i] - mx) * inv;
  attn_acc[tid] = 0.0f;  // blockDim == H
}

// ---------- attn_applied[c] = sum_j w[j] * enc[j, c]  (split-K, float atomics) ----------
__global__ void attn_apply_kernel(const float* __restrict__ enc, const float* __restrict__ w,
                                  float* __restrict__ acc) {
  int col = blockIdx.x * blockDim.x + threadIdx.x;  // gridDim.x=4, blockDim=256 -> 0..1023
  int j0 = blockIdx.y * 128;
  float s = 0.f;
  for (int j = j0; j < j0 + 128; ++j)
    s += w[j] * enc[(size_t)j * H + col];
  atomicAdd(&acc[col], s);
}

// ---------- GRU gate math (PyTorch order r,z,n) ----------
__global__ void gru_kernel(const float* __restrict__ gi, const float* __restrict__ gh,
                           const float* __restrict__ h0, float* __restrict__ hnew) {
  int t = blockIdx.x * blockDim.x + threadIdx.x;
  if (t >= H) return;
  float r = 1.f / (1.f + expf(-(gi[t] + gh[t])));
  float z = 1.f / (1.f + expf(-(gi[H + t] + gh[H + t])));
  float n = tanhf(gi[2 * H + t] + r * gh[2 * H + t]);
  hnew[t] = (1.f - z) * n + z * h0[t];
}

// ---------- log-softmax over V logits: 3 small passes ----------
__global__ void blockmax_kernel(const float* __restrict__ l, float* __restrict__ bmax) {
  __shared__ float red[512];
  int i = blockIdx.x * 512 + threadIdx.x;
  red[threadIdx.x] = l[i];
  __syncthreads();
  for (int s = 256; s > 0; s >>= 1) {
    if (threadIdx.x < s) red[threadIdx.x] = fmaxf(red[threadIdx.x], red[threadIdx.x + s]);
    __syncthreads();
  }
  if (threadIdx.x == 0) bmax[blockIdx.x] = red[0];
}

__global__ void finalmax_kernel(const float* __restrict__ bmax, int nb, float* __restrict__ mx_sum) {
  __shared__ float red[256];
  float m = -1e30f;
  for (int i = threadIdx.x; i < nb; i += 256) m = fmaxf(m, bmax[i]);
  red[threadIdx.x] = m; __syncthreads();
  for (int s = 128; s > 0; s >>= 1) {
    if (threadIdx.x < s) red[threadIdx.x] = fmaxf(red[threadIdx.x], red[threadIdx.x + s]);
    __syncthreads();
  }
  if (threadIdx.x == 0) { mx_sum[0] = red[0]; mx_sum[1] = 0.0f; }
}

__global__ void sumexp_kernel(const float* __restrict__ l, float* __restrict__ mx_sum) {
  __shared__ float red[512];
  float mx = mx_sum[0];
  int i = blockIdx.x * 512 + threadIdx.x;
  red[threadIdx.x] = expf(l[i] - mx);
  __syncthreads();
  for (int s = 256; s > 0; s >>= 1) {
    if (threadIdx.x < s) red[threadIdx.x] += red[threadIdx.x + s];
    __syncthreads();
  }
  if (threadIdx.x == 0) atomicAdd(&mx_sum[1], red[0]);
}

__global__ void logsoftmax_kernel(float* __restrict__ l, const float* __restrict__ mx_sum) {
  int i = blockIdx.x * blockDim.x + threadIdx.x;
  float c = mx_sum[0] + logf(mx_sum[1]);
  l[i] = l[i] - c;
}

extern "C" void kernel_launch(void* const* d_in, const int* in_sizes, int n_in,
                              void* d_out, int out_size, void* d_ws, size_t ws_size,
                              hipStream_t stream) {
  const int*   ids    = (const int*)d_in[0];
  const float* hidden = (const float*)d_in[1];
  const float* enc    = (const float*)d_in[2];
  const float* emb    = (const float*)d_in[3];
  const float* attn_w = (const float*)d_in[4];
  const float* attn_b = (const float*)d_in[5];
  const float* comb_w = (const float*)d_in[6];
  const float* comb_b = (const float*)d_in[7];
  const float* w_ih   = (const float*)d_in[8];
  const float* w_hh   = (const float*)d_in[9];
  const float* b_ih   = (const float*)d_in[10];
  const float* b_hh   = (const float*)d_in[11];
  const float* out_w  = (const float*)d_in[12];
  const float* out_b  = (const float*)d_in[13];

  float* out      = (float*)d_out;  // [V] log-probs (normalized in place)
  float* h_new    = out + V;        // [H]
  float* attn_wts = out + V + H;    // [ML]

  float* ws      = (float*)d_ws;
  float* xcat1   = ws;              // 2048  [e0, h0]
  float* xcat2   = ws + 2048;       // 2048  [e0, attn_applied]
  float* alogits = ws + 4096;       // 4096
  float* xrelu   = ws + 8192;       // 1024
  float* gi      = ws + 9216;       // 3072
  float* gh      = ws + 12288;      // 3072
  float* bmax    = ws + 15360;      // 256
  float* mx_sum  = ws + 15616;      // 2

  prep_kernel<<<8, 256, 0, stream>>>(ids, hidden, emb, xcat1, xcat2);
  wmma_gemv<false><<<ML / 128, 256, 0, stream>>>(attn_w, xcat1, attn_b, alogits, 2 * H, 0);
  softmax_kernel<<<1, 1024, 0, stream>>>(alogits, attn_wts, xcat2 + H);
  attn_apply_kernel<<<dim3(4, 32), 256, 0, stream>>>(enc, attn_wts, xcat2 + H);
  wmma_gemv<false><<<H / 128, 256, 0, stream>>>(comb_w, xcat2, comb_b, xrelu, 2 * H, 1);
  wmma_gemv<false><<<(3 * H) / 128, 256, 0, stream>>>(w_ih, xrelu, b_ih, gi, H, 0);
  wmma_gemv<false><<<(3 * H) / 128, 256, 0, stream>>>(w_hh, hidden, b_hh, gh, H, 0);
  gru_kernel<<<4, 256, 0, stream>>>(gi, gh, hidden, h_new);
  // Dominant kernel: 512 MB stream, non-temporal so the ~80 MB of other weights stay in L2.
  wmma_gemv<true><<<V / 128, 256, 0, stream>>>(out_w, h_new, out_b, out, H, 0);
  blockmax_kernel<<<V / 512, 512, 0, stream>>>(out, bmax);
  finalmax_kernel<<<1, 256, 0, stream>>>(bmax, V / 512, mx_sum);
  sumexp_kernel<<<V / 512, 512, 0, stream>>>(out, mx_sum);
  logsoftmax_kernel<<<V / 256, 256, 0, stream>>>(out, mx_sum);
}